// QuantumGate_65481071403710
// MI455X (gfx1250) — compile-verified
//
#include <hip/hip_runtime.h>
#include <stdint.h>

// out[b,w] = prod_{k<=w} cos(inputs[b,k])
// Analytic reduction: product-state prep + CNOT chain => U† Z_w U = Z_0..Z_w,
// <Z_k> of RZ(phi)RX(theta)|0> = cos(theta); RZ params provably irrelevant.

#define RPB 256   // rows per block (8 wave32s)
#define NW  10    // wires

__global__ __launch_bounds__(RPB) void qsim_prefix_cos_kernel(
    const float* __restrict__ in, float* __restrict__ out, int B)
{
    __shared__ float s_in [RPB * NW];
    __shared__ float s_out[RPB * NW];

    const int       t     = threadIdx.x;
    const long long flat0 = (long long)blockIdx.x * (RPB * NW);
    const long long nflat = (long long)B * NW;
    const bool      full  = (flat0 + (long long)(RPB * NW)) <= nflat; // block-uniform

    const uint64_t gin  = (uint64_t)(uintptr_t)(in  + flat0);
    const unsigned goff = (unsigned)(t * 4);                  // byte offset of lane's first elem
    const unsigned lin  = (unsigned)(uintptr_t)(&s_in[t]);    // LDS byte addr, same layout

    // ---- Phase 1: global -> LDS, coalesced async DMA. INST_OFFSET (= m*1024B)
    // is added to BOTH the LDS and global addresses, so one addr VGPR suffices.
    if (full) {
        #pragma unroll
        for (int m = 0; m < NW; ++m) {
            asm volatile("global_load_async_to_lds_b32 %0, %1, %2 offset:%3"
                         :: "v"(lin), "v"(goff), "s"(gin), "i"(m * RPB * 4)
                         : "memory");
        }
    } else {
        #pragma unroll
        for (int m = 0; m < NW; ++m) {
            const int j = t + m * RPB;
            if (flat0 + j < nflat) {
                asm volatile("global_load_async_to_lds_b32 %0, %1, %2 offset:%3"
                             :: "v"(lin), "v"(goff), "s"(gin), "i"(m * RPB * 4)
                             : "memory");
            }
        }
    }
    asm volatile("s_wait_asynccnt 0" ::: "memory");
    __syncthreads();

    // ---- Phase 2: per-row prefix product of cosines.
    // Stride-10 LDS reads: 10t mod 64 hits 32 distinct banks per wave => conflict-free.
    const int row = blockIdx.x * RPB + t;
    if (row < B) {
        float p = 1.0f;
        #pragma unroll
        for (int w = 0; w < NW; ++w) {
            p *= __cosf(s_in[t * NW + w]);   // v_cos_f32; inputs ~N(0,1), tiny arg range
            s_out[t * NW + w] = p;
        }
    }
    __syncthreads();

    // ---- Phase 3: LDS -> global, coalesced async DMA store.
    const uint64_t gout = (uint64_t)(uintptr_t)(out + flat0);
    const unsigned lout = (unsigned)(uintptr_t)(&s_out[t]);
    if (full) {
        #pragma unroll
        for (int m = 0; m < NW; ++m) {
            asm volatile("global_store_async_from_lds_b32 %0, %1, %2 offset:%3"
                         :: "v"(goff), "v"(lout), "s"(gout), "i"(m * RPB * 4)
                         : "memory");
        }
    } else {
        #pragma unroll
        for (int m = 0; m < NW; ++m) {
            const int j = t + m * RPB;
            if (flat0 + j < nflat) {
                asm volatile("global_store_async_from_lds_b32 %0, %1, %2 offset:%3"
                             :: "v"(goff), "v"(lout), "s"(gout), "i"(m * RPB * 4)
                             : "memory");
            }
        }
    }
    asm volatile("s_wait_asynccnt 0" ::: "memory");
    // s_endpgm's implicit wait-idle also covers any outstanding ASYNCcnt.
}

extern "C" void kernel_launch(void* const* d_in, const int* in_sizes, int n_in,
                              void* d_out, int out_size, void* d_ws, size_t ws_size,
                              hipStream_t stream) {
    const float* inputs = (const float*)d_in[0];
    // d_in[1] (rz_params) is analytically irrelevant to <Z_w>; intentionally unused.
    (void)d_ws; (void)ws_size; (void)n_in; (void)out_size;

    const int B = in_sizes[0] / NW;           // 8192
    const int blocks = (B + RPB - 1) / RPB;   // 32
    qsim_prefix_cos_kernel<<<blocks, RPB, 0, stream>>>(inputs, (float*)d_out, B);
}